// GMM_78649441124919
// MI455X (gfx1250) — compile-verified
//
#include <hip/hip_runtime.h>
#include <hip/hip_bf16.h>
#include <math.h>

typedef __attribute__((ext_vector_type(2))) float v2f;
typedef __attribute__((ext_vector_type(8))) float v8f;

#define GMM_D 8
#define GMM_K 16
#define GMM_F 48          // 36 quad + 8 linear + 1 const + 3 pad
#define GMM_EPS 1e-6f
#define LOG2PI 1.8378770664093453f

// feature f is xe[FU[f]] * xe[FV[f]], where xe = {x0..x7, 1.0f, 0.0f}
// f 0..35  : x_i * x_j, i<=j, enumerated (i outer, j inner)
// f 36..43 : x_i * 1
// f 44     : 1 * 1      (constant term)
// f 45..47 : 0 * 0      (padding)
constexpr int FU[GMM_F] = {
    0,0,0,0,0,0,0,0,
    1,1,1,1,1,1,1,
    2,2,2,2,2,2,
    3,3,3,3,3,
    4,4,4,4,
    5,5,5,
    6,6,
    7,
    0,1,2,3,4,5,6,7,
    8, 9,9,9};
constexpr int FV[GMM_F] = {
    0,1,2,3,4,5,6,7,
    1,2,3,4,5,6,7,
    2,3,4,5,6,7,
    3,4,5,6,7,
    4,5,6,7,
    5,6,7,
    6,7,
    7,
    8,8,8,8,8,8,8,8,
    8, 9,9,9};

// ---------------------------------------------------------------------------
// Setup: one wave, lane k (k<16) builds the 48-entry weight row for component k.
//   cov = tril(Lc) tril(Lc)^T + eps I ; L = chol(cov) ; A = inv(L) ; P = A^T A
//   score(x) = -0.5 x^T P x + (P mu)^T x + const_k
// ---------------------------------------------------------------------------
__global__ void GMM_setup_kernel(const float* __restrict__ means,
                                 const float* __restrict__ chol,
                                 const float* __restrict__ pi,
                                 float* __restrict__ W) {
    int k = threadIdx.x;
    if (k >= GMM_K) return;

    float mx = -INFINITY;
    for (int i = 0; i < GMM_K; ++i) mx = fmaxf(mx, pi[i]);
    float s = 0.f;
    for (int i = 0; i < GMM_K; ++i) s += expf(pi[i] - mx);
    float logsoft = pi[k] - (mx + logf(s));

    float Lc[GMM_D][GMM_D];
    for (int i = 0; i < GMM_D; ++i)
        for (int j = 0; j < GMM_D; ++j)
            Lc[i][j] = (j <= i) ? chol[k * 64 + i * 8 + j] : 0.f;

    float cov[GMM_D][GMM_D];
    for (int i = 0; i < GMM_D; ++i)
        for (int j = 0; j < GMM_D; ++j) {
            float acc = 0.f;
            for (int t = 0; t < GMM_D; ++t) acc += Lc[i][t] * Lc[j][t];
            cov[i][j] = acc + ((i == j) ? GMM_EPS : 0.f);
        }

    float L[GMM_D][GMM_D];
    for (int i = 0; i < GMM_D; ++i)
        for (int j = 0; j < GMM_D; ++j) L[i][j] = 0.f;
    for (int i = 0; i < GMM_D; ++i)
        for (int j = 0; j <= i; ++j) {
            float acc = cov[i][j];
            for (int t = 0; t < j; ++t) acc -= L[i][t] * L[j][t];
            if (i == j) L[i][j] = sqrtf(acc);
            else        L[i][j] = acc / L[j][j];
        }
    float half_logdet = 0.f;
    for (int i = 0; i < GMM_D; ++i) half_logdet += logf(L[i][i]);

    float A[GMM_D][GMM_D];
    for (int i = 0; i < GMM_D; ++i)
        for (int j = 0; j < GMM_D; ++j) A[i][j] = 0.f;
    for (int c = 0; c < GMM_D; ++c)
        for (int r = c; r < GMM_D; ++r) {
            float acc = (r == c) ? 1.f : 0.f;
            for (int t = c; t < r; ++t) acc -= L[r][t] * A[t][c];
            A[r][c] = acc / L[r][r];
        }

    float P[GMM_D][GMM_D];
    for (int i = 0; i < GMM_D; ++i)
        for (int j = 0; j < GMM_D; ++j) {
            float acc = 0.f;
            for (int r = 0; r < GMM_D; ++r) acc += A[r][i] * A[r][j];
            P[i][j] = acc;
        }

    float mu[GMM_D];
    for (int i = 0; i < GMM_D; ++i) mu[i] = means[k * GMM_D + i];
    float Pmu[GMM_D];
    float muPmu = 0.f;
    for (int i = 0; i < GMM_D; ++i) {
        float acc = 0.f;
        for (int j = 0; j < GMM_D; ++j) acc += P[i][j] * mu[j];
        Pmu[i] = acc;
        muPmu += mu[i] * acc;
    }

    float* w = W + k * GMM_F;
    int idx = 0;
    for (int i = 0; i < GMM_D; ++i)
        for (int j = i; j < GMM_D; ++j)
            w[idx++] = -0.5f * P[i][j] * ((i == j) ? 1.f : 2.f);
    for (int i = 0; i < GMM_D; ++i) w[36 + i] = Pmu[i];
    w[44] = -0.5f * muPmu - 0.5f * GMM_D * LOG2PI - half_logdet + logsoft;
    w[45] = 0.f; w[46] = 0.f; w[47] = 0.f;
}

// ---------------------------------------------------------------------------
// Main: one wave per 16-point tile.
//   A (16x4)  = W slice      -> D rows   = components
//   B (4x16)  = Z^T features -> D cols   = points
// D layout: lane p (0..15) holds comps 0..7 in acc[0..7]; lane p+16 holds
// comps 8..15 for the same point -> logsumexp is 7 in-lane fmax + 1 shuffle,
// 8 exps + 1 shuffle, 1 log. Lanes 0..15 store coalesced.
// ---------------------------------------------------------------------------
__global__ __launch_bounds__(256) void GMM_main_kernel(
        const float* __restrict__ x,
        const float* __restrict__ W,
        float* __restrict__ out,
        int n) {
    const int lane = threadIdx.x & 31;
    const int wave = blockIdx.x * (blockDim.x >> 5) + (threadIdx.x >> 5);
    const int base = wave * 16;
    if (base >= n) return;                 // wave-uniform exit, EXEC stays full

    const bool hi = lane >= 16;
    const int  m  = lane & 15;             // A: component row ; B: point col
    const int  pt = min(base + m, n - 1);  // clamp keeps EXEC all-ones

    // lane's point coords + {1, 0} feature sources (constant indices only)
    const float4 x0 = *(const float4*)(x + (size_t)pt * GMM_D);
    const float4 x1 = *(const float4*)(x + (size_t)pt * GMM_D + 4);
    float xe[10] = {x0.x, x0.y, x0.z, x0.w, x1.x, x1.y, x1.z, x1.w, 1.0f, 0.0f};

    // A operands: lane holds W[comp=m][4s + 2*hi], W[comp=m][4s + 2*hi + 1]
    const float* wp = W + m * GMM_F + (hi ? 2 : 0);
    v2f aop[12];
#pragma unroll
    for (int s = 0; s < 12; ++s)
        aop[s] = *(const v2f*)(wp + 4 * s);

    v8f acc = {};
#pragma unroll
    for (int s = 0; s < 12; ++s) {
        // B operand: 2 features of this lane's point.
        // Compute BOTH half-candidates as products (literal-constant array
        // indices -> pure registers), then select between the VALUES so the
        // compiler emits one v_cndmask each instead of a dynamic-index tree.
        const int f0 = 4 * s, f1 = 4 * s + 1, f2 = 4 * s + 2, f3 = 4 * s + 3;
        const float plx = xe[FU[f0]] * xe[FV[f0]];
        const float ply = xe[FU[f1]] * xe[FV[f1]];
        const float phx = xe[FU[f2]] * xe[FV[f2]];
        const float phy = xe[FU[f3]] * xe[FV[f3]];
        v2f b;
        b.x = hi ? phx : plx;
        b.y = hi ? phy : ply;
        acc = __builtin_amdgcn_wmma_f32_16x16x4_f32(
                  /*neg_a=*/false, aop[s], /*neg_b=*/false, b,
                  /*c_mod=*/(short)0, acc,
                  /*reuse_a=*/false, /*reuse_b=*/false);
    }

    // ---- logsumexp over 16 components: 8 in-lane + cross-half partner ----
    float mxv = acc[0];
#pragma unroll
    for (int v = 1; v < 8; ++v) mxv = fmaxf(mxv, acc[v]);
    mxv = fmaxf(mxv, __shfl_xor(mxv, 16, 32));   // max over all 16 comps

    float sm = 0.f;
#pragma unroll
    for (int v = 0; v < 8; ++v) sm += __expf(acc[v] - mxv);
    sm += __shfl_xor(sm, 16, 32);                // sum over all 16 comps

    const float res = mxv + __logf(sm);

    const int p = base + m;
    if (!hi && p < n) out[p] = res;              // coalesced 64B store
}

extern "C" void kernel_launch(void* const* d_in, const int* in_sizes, int n_in,
                              void* d_out, int out_size, void* d_ws, size_t ws_size,
                              hipStream_t stream) {
    const float* x     = (const float*)d_in[0];   // [N,8]
    const float* means = (const float*)d_in[1];   // [16,8]
    const float* chol  = (const float*)d_in[2];   // [16,8,8]
    const float* pi    = (const float*)d_in[3];   // [16]
    float* out = (float*)d_out;
    float* W   = (float*)d_ws;                    // [16,48] = 3 KB scratch

    const int n = in_sizes[0] / GMM_D;

    GMM_setup_kernel<<<1, 32, 0, stream>>>(means, chol, pi, W);

    const int tiles = (n + 15) / 16;              // one wave per tile
    const int waves_per_block = 8;                // 256 threads
    const int blocks = (tiles + waves_per_block - 1) / waves_per_block;
    GMM_main_kernel<<<blocks, 32 * waves_per_block, 0, stream>>>(x, W, out, n);
}